// CubicalLayer_3393024164456
// MI455X (gfx1250) — compile-verified
//
#include <hip/hip_runtime.h>

// CubicalLayer gather for MI455X (gfx1250, wave32).
//
// out[i] = X[indices[i,0], indices[i,1]]  for i in [0, 524288)
// X: 4096 x 4096 fp32 (64 MB -> fully L2-resident on the 192 MB L2).
// Pure gather, 0 FLOPs: memory-latency bound. Optimize MLP, coalescing,
// 32-bit address math, and cache policy. WMMA is inapplicable (no matmul).

typedef int   v4i __attribute__((ext_vector_type(4)));
typedef float v2f __attribute__((ext_vector_type(2)));

// Row stride fixed by the reference (W = 4096). Element offset (row<<12)+col
// is < 2^24, so all gather addressing fits in 32-bit offsets off a 64-bit
// scalar base -> single v_lshl_add_u32 per gather instead of 64-bit chains.
static constexpr int LOG2_W = 12;

__global__ __launch_bounds__(256) void cubical_gather2_kernel(
    const float* __restrict__ X,
    const int*   __restrict__ idx,
    float*       __restrict__ out,
    int n_pairs)                       // even
{
    const unsigned t    = blockIdx.x * blockDim.x + threadIdx.x;
    const unsigned base = t << 1;                  // two (row,col) pairs per thread
    if (base >= (unsigned)n_pairs) return;

    // 16B coalesced load: two index pairs per lane (512 B per wave32).
    const v4i p = *reinterpret_cast<const v4i*>(idx + (size_t)base * 2u);

    // Keep the streaming index read-front ahead of use.
    // Lowers to global_prefetch_b8 (gfx1250). Clamp so the address is
    // always valid (WGP-scope prefetch is treated as non-speculative).
    {
        unsigned pf   = base * 2u + 8192u;         // 32 KB ahead
        unsigned last = (unsigned)n_pairs * 2u - 1u;
        __builtin_prefetch(idx + (pf < last ? pf : last), 0, 3);
    }

    // 32-bit element offsets -> v_lshl_add_u32 address math.
    const unsigned off0 = ((unsigned)p.x << LOG2_W) + (unsigned)p.y;
    const unsigned off1 = ((unsigned)p.z << LOG2_W) + (unsigned)p.w;

    // Two independent random gathers -> 2x memory-level parallelism per lane.
    // Default (RT) policy keeps the 64 MB matrix resident in the 192 MB L2
    // across graph replays.
    const float v0 = X[off0];
    const float v1 = X[off1];

    // Coalesced 8B non-temporal store (TH=NT): output is write-once, read
    // back by the host -- don't let it evict X from L2.
    v2f o;
    o.x = v0;
    o.y = v1;
    __builtin_nontemporal_store(o, reinterpret_cast<v2f*>(out + base));
}

// Scalar fallback for odd counts (never hit for N_IDX = 524288).
__global__ __launch_bounds__(256) void cubical_gather1_kernel(
    const float* __restrict__ X,
    const int*   __restrict__ idx,
    float*       __restrict__ out,
    int n_pairs)
{
    const unsigned i = blockIdx.x * blockDim.x + threadIdx.x;
    if (i >= (unsigned)n_pairs) return;
    const int r = idx[(size_t)i * 2u];
    const int c = idx[(size_t)i * 2u + 1u];
    const unsigned off = ((unsigned)r << LOG2_W) + (unsigned)c;
    __builtin_nontemporal_store(X[off], out + i);
}

extern "C" void kernel_launch(void* const* d_in, const int* in_sizes, int n_in,
                              void* d_out, int out_size, void* d_ws, size_t ws_size,
                              hipStream_t stream)
{
    const float* X   = (const float*)d_in[0];   // 4096*4096 fp32
    const int*   idx = (const int*)d_in[1];     // 524288*2 int32
    float*       out = (float*)d_out;           // 524288 fp32

    const int n_pairs = in_sizes[1] / 2;        // 524288
    const int block   = 256;                    // 8 wave32 per block

    if ((n_pairs & 1) == 0) {
        const int threads = n_pairs / 2;        // 2 gathers per thread
        const int grid    = (threads + block - 1) / block;
        cubical_gather2_kernel<<<grid, block, 0, stream>>>(X, idx, out, n_pairs);
    } else {
        const int grid = (n_pairs + block - 1) / block;
        cubical_gather1_kernel<<<grid, block, 0, stream>>>(X, idx, out, n_pairs);
    }
}